// gumbel_softmax_46437186404426
// MI455X (gfx1250) — compile-verified
//
#include <hip/hip_runtime.h>
#include <hip/hip_bf16.h>
#include <stdint.h>

// Reference collapses to: out[i] = 1 for all i (softmax over a singleton axis
// is identically 1, argmax of a singleton is 0, branch -> +1). Inputs are
// mathematically irrelevant, so the optimal MI455X kernel is a pure store
// stream: 134 MB of writes, ~5.8 us at 23.3 TB/s, zero reads, zero FLOPs.
// No matrix math exists -> WMMA inapplicable; the gfx1250 path used instead is
// the async LDS->global store engine (ASYNCcnt) with ds_store_b128-staged
// constants, falling back to plain global_store_b128 if the builtins are
// absent in this toolchain.

#define AS_GLOBAL __attribute__((address_space(1)))
#define AS_LDS    __attribute__((address_space(3)))

// Pointee type per hipcc diagnostic: __attribute__((vector_size(16))) int
typedef int v4i __attribute__((vector_size(16)));

__global__ void gumbel_const_fill_kernel(float* __restrict__ out,
                                         long long n4, long long n) {
  __shared__ float4 tile[256];  // one 16B constant slot per thread (4 KB LDS)
  const int t = threadIdx.x;
  tile[t] = make_float4(1.0f, 1.0f, 1.0f, 1.0f);   // ds_store_b128
  __syncthreads();                                  // DS write visible before async engine reads

  float4* __restrict__ out4 = reinterpret_cast<float4*>(out);
  const long long idx    = (long long)blockIdx.x * blockDim.x + t;
  const long long stride = (long long)gridDim.x * (long long)blockDim.x;

#if defined(__gfx1250__) && __has_builtin(__builtin_amdgcn_global_store_async_from_lds_b128)
  // Async store engine: per-lane 16B LDS -> global, tracked by ASYNCcnt.
  // Data never touches VGPRs; the wave just issues addresses.
  AS_LDS v4i* lsrc = (AS_LDS v4i*)(uint32_t)(uintptr_t)&tile[t];
  for (long long i = idx; i < n4; i += stride) {
    AS_GLOBAL v4i* gdst = (AS_GLOBAL v4i*)(uintptr_t)(out4 + i);
    __builtin_amdgcn_global_store_async_from_lds_b128(gdst, lsrc, /*offset=*/0, /*cpol=*/0);
  }
#if __has_builtin(__builtin_amdgcn_s_wait_asynccnt)
  __builtin_amdgcn_s_wait_asynccnt(0);              // s_wait_asynccnt 0
#endif
#else
  // Fallback: plain wide stores -> global_store_b128, still HBM-saturating.
  const float4 v = make_float4(1.0f, 1.0f, 1.0f, 1.0f);
  for (long long i = idx; i < n4; i += stride) out4[i] = v;
#endif

  // Scalar tail (out_size is divisible by 4 here, kept for generality).
  for (long long i = n4 * 4 + idx; i < n; i += stride) out[i] = 1.0f;
}

extern "C" void kernel_launch(void* const* d_in, const int* in_sizes, int n_in,
                              void* d_out, int out_size, void* d_ws, size_t ws_size,
                              hipStream_t stream) {
  (void)d_in; (void)in_sizes; (void)n_in; (void)d_ws; (void)ws_size;
  float* out = (float*)d_out;                 // non-bf16 output -> float* per harness
  const long long n  = (long long)out_size;   // 2048*16384 = 33,554,432
  const long long n4 = n >> 2;                // 8,388,608 x 16B stores
  const int  threads = 256;                   // 8 wave32 per block
  const int  blocks  = 4096;                  // ~8 B128 stores per thread
  gumbel_const_fill_kernel<<<blocks, threads, 0, stream>>>(out, n4, n);
}